// GraphSAGENet_90546500535018
// MI455X (gfx1250) — compile-verified
//
#include <hip/hip_runtime.h>
#include <cstddef>

typedef __attribute__((ext_vector_type(2))) float v2f;
typedef __attribute__((ext_vector_type(8))) float v8f;

#define NNODES 50000
#define NEDGES 800000
#define CHID   128
#define BN_EPS 1e-5f

// ---------------------------------------------------------------- utilities
__global__ void fill_kernel(float* __restrict__ p, float v, size_t n) {
    size_t i = (size_t)blockIdx.x * blockDim.x + threadIdx.x;
    if (i < n) p[i] = v;
}

__global__ void degree_kernel(const int* __restrict__ ei, float* __restrict__ deg, int E) {
    int e = blockIdx.x * blockDim.x + threadIdx.x;
    if (e >= E) return;
    int dst = ei[E + e];
    unsafeAtomicAdd(&deg[dst], 1.0f);
}

__global__ void invdeg_kernel(float* __restrict__ deg, int n) {
    int i = blockIdx.x * blockDim.x + threadIdx.x;
    if (i >= n) return;
    float d = deg[i];
    deg[i] = (d > 0.0f) ? (1.0f / d) : 0.0f;
}

// one wave32 per edge; C=128 channels -> 4 per lane (float4 gather)
__global__ void aggregate_kernel(const float* __restrict__ h, const int* __restrict__ ei,
                                 const float* __restrict__ ew, float* __restrict__ agg, int E) {
    int e = blockIdx.x * (blockDim.x >> 5) + (threadIdx.x >> 5);
    if (e >= E) return;
    int lane = threadIdx.x & 31;
    int src = ei[e];
    int dst = ei[E + e];
    float w = ew[e];
    float4 v = ((const float4*)(h + (size_t)src * CHID))[lane];
    float* a = agg + (size_t)dst * CHID + lane * 4;
    unsafeAtomicAdd(a + 0, v.x * w);
    unsafeAtomicAdd(a + 1, v.y * w);
    unsafeAtomicAdd(a + 2, v.z * w);
    unsafeAtomicAdd(a + 3, v.w * w);
}

// ------------------------------------------------- WMMA f32 GEMM (+epilogue)
// out[16 x COUT] tile per block (one 16x16 tile per wave), K fully unrolled.
// out = epilogue( A1*rowScale @ W1 [+ A2 @ W2] + bias ), epilogue = relu, +resid,
// and per-channel sum/sumsq accumulation for BN.
template <int K, int COUT, bool HAS_A2, bool RELU, bool STATS>
__global__ __launch_bounds__(32 * (COUT / 16)) void sage_gemm(
    const float* __restrict__ A1, const float* __restrict__ rowScale,
    const float* __restrict__ W1,
    const float* __restrict__ A2, const float* __restrict__ W2,
    const float* __restrict__ bias, const float* __restrict__ resid,
    float* __restrict__ out, float* __restrict__ statSum, float* __restrict__ statSq) {
    constexpr int LDA = K + 4;               // pad: kill stride-K LDS bank conflicts
    __shared__ float lA[16 * LDA];
    __shared__ float lB[HAS_A2 ? 16 * LDA : 16];

    const int row0 = blockIdx.x * 16;
    const int tid  = threadIdx.x;
    const int NT   = 32 * (COUT / 16);

    for (int i = tid; i < 16 * K; i += NT) {
        int r = i / K, c = i % K;
        float v = A1[(size_t)(row0 + r) * K + c];
        if (rowScale) v *= rowScale[row0 + r];
        lA[r * LDA + c] = v;
        if (HAS_A2) lB[r * LDA + c] = A2[(size_t)(row0 + r) * K + c];
    }
    __syncthreads();

    const int lane = tid & 31;
    const int half = lane >> 4;   // 0: lanes 0-15 (K=k,k+1 / M=v), 1: lanes 16-31 (K=k+2,k+3 / M=v+8)
    const int lm   = lane & 15;
    const int col0 = (tid >> 5) << 4;
    const int col  = col0 + lm;

    v8f acc = {0.f, 0.f, 0.f, 0.f, 0.f, 0.f, 0.f, 0.f};
#pragma unroll
    for (int k = 0; k < K; k += 4) {
        int ka = k + 2 * half;
        v2f a, b;
        a.x = lA[lm * LDA + ka];
        a.y = lA[lm * LDA + ka + 1];
        b.x = W1[(size_t)ka * COUT + col];
        b.y = W1[(size_t)(ka + 1) * COUT + col];
        acc = __builtin_amdgcn_wmma_f32_16x16x4_f32(false, a, false, b, (short)0, acc, false, false);
    }
    if (HAS_A2) {
#pragma unroll
        for (int k = 0; k < K; k += 4) {
            int ka = k + 2 * half;
            v2f a, b;
            a.x = lB[lm * LDA + ka];
            a.y = lB[lm * LDA + ka + 1];
            b.x = W2[(size_t)ka * COUT + col];
            b.y = W2[(size_t)(ka + 1) * COUT + col];
            acc = __builtin_amdgcn_wmma_f32_16x16x4_f32(false, a, false, b, (short)0, acc, false, false);
        }
    }

    const float bv = bias[col];
    float s = 0.f, sq = 0.f;
#pragma unroll
    for (int v = 0; v < 8; ++v) {
        int m = v + 8 * half;
        float val = acc[v] + bv;
        if (RELU) val = fmaxf(val, 0.f);
        if (resid) val += resid[(size_t)(row0 + m) * COUT + col];
        out[(size_t)(row0 + m) * COUT + col] = val;
        s += val;
        sq += val * val;
    }
    if (STATS) {
        unsafeAtomicAdd(&statSum[col], s);
        unsafeAtomicAdd(&statSq[col], sq);
    }
}

// ------------------------------------------------------------------- BN
__global__ void bn_finalize_kernel(const float* __restrict__ ssum, const float* __restrict__ ssq,
                                   const float* __restrict__ g, const float* __restrict__ be,
                                   float* __restrict__ scale, float* __restrict__ shift, int C) {
    int c = threadIdx.x;
    if (c >= C) return;
    float mu  = ssum[c] / (float)NNODES;
    float var = ssq[c] / (float)NNODES - mu * mu;
    float s   = g[c] * rsqrtf(var + BN_EPS);
    scale[c] = s;
    shift[c] = be[c] - mu * s;
}

__global__ void bn_apply_kernel(const float* __restrict__ in, const float* __restrict__ scale,
                                const float* __restrict__ shift, float* __restrict__ out,
                                size_t n, int Cmask) {
    size_t i = (size_t)blockIdx.x * blockDim.x + threadIdx.x;
    if (i >= n) return;
    int c = (int)(i & (size_t)Cmask);
    out[i] = in[i] * scale[c] + shift[c];
}

// ------------------------------------------------------------------ driver
extern "C" void kernel_launch(void* const* d_in, const int* in_sizes, int n_in,
                              void* d_out, int out_size, void* d_ws, size_t ws_size,
                              hipStream_t stream) {
    const float* x  = (const float*)d_in[0];
    const int*   ei = (const int*)d_in[1];     // int32 (JAX x64 disabled)
    const float* ew = (const float*)d_in[2];
    const float* Wl[4] = {(const float*)d_in[3], (const float*)d_in[6], (const float*)d_in[9],  (const float*)d_in[12]};
    const float* Wr[4] = {(const float*)d_in[4], (const float*)d_in[7], (const float*)d_in[10], (const float*)d_in[13]};
    const float* bb[4] = {(const float*)d_in[5], (const float*)d_in[8], (const float*)d_in[11], (const float*)d_in[14]};
    const float* fcW  = (const float*)d_in[15]; const float* fcB  = (const float*)d_in[16];
    const float* fc1W = (const float*)d_in[17]; const float* fc1B = (const float*)d_in[18];
    const float* fc2W = (const float*)d_in[19]; const float* fc2B = (const float*)d_in[20];
    const float* g[5]  = {(const float*)d_in[21], (const float*)d_in[23], (const float*)d_in[25],
                          (const float*)d_in[27], (const float*)d_in[29]};
    const float* be[5] = {(const float*)d_in[22], (const float*)d_in[24], (const float*)d_in[26],
                          (const float*)d_in[28], (const float*)d_in[30]};

    float* ws   = (float*)d_ws;
    float* agg  = ws;                                   // N*128, also pre-BN buffer t
    float* hA   = agg  + (size_t)NNODES * CHID;         // N*128
    float* hB   = hA   + (size_t)NNODES * CHID;         // N*128
    float* invd = hB   + (size_t)NNODES * CHID;         // N
    float* ssum = invd + NNODES;                        // 128
    float* ssq  = ssum + 128;                           // 128
    float* sc   = ssq  + 128;                           // 128
    float* sh   = sc   + 128;                           // 128

    auto fill = [&](float* p, size_t n, float v) {
        fill_kernel<<<(int)((n + 255) / 256), 256, 0, stream>>>(p, v, n);
    };

    // degrees -> inv_deg (in place)
    fill(invd, NNODES, 0.f);
    degree_kernel<<<(NEDGES + 255) / 256, 256, 0, stream>>>(ei, invd, NEDGES);
    invdeg_kernel<<<(NNODES + 255) / 256, 256, 0, stream>>>(invd, NNODES);

    const int GB = NNODES / 16;  // 3125 tile rows, exact
    const size_t NC  = (size_t)NNODES * CHID;
    const size_t NC2 = (size_t)NNODES * 64;

    // ---- SAGE layers 1..4 ----
    const float* hin = x;
    float* houts[3] = {hA, hB, hA};  // post-BN outputs of layers 1..3; layer4 stays in agg
    for (int L = 0; L < 4; ++L) {
        fill(agg, NC, 0.f);
        aggregate_kernel<<<(NEDGES + 7) / 8, 256, 0, stream>>>(hin, ei, ew, agg, NEDGES);
        if (L < 3) {
            fill(ssum, 256, 0.f);
            sage_gemm<128, 128, true, true, true><<<GB, 256, 0, stream>>>(
                agg, invd, Wl[L], hin, Wr[L], bb[L], (L == 0) ? nullptr : hin,
                agg, ssum, ssq);
            bn_finalize_kernel<<<1, 128, 0, stream>>>(ssum, ssq, g[L], be[L], sc, sh, 128);
            bn_apply_kernel<<<(int)((NC + 255) / 256), 256, 0, stream>>>(agg, sc, sh, houts[L], NC, 127);
            hin = houts[L];
        } else {
            // x4 = relu(sage(x3)) + x3  (no BN) -> stays in agg
            sage_gemm<128, 128, true, true, false><<<GB, 256, 0, stream>>>(
                agg, invd, Wl[L], hin, Wr[L], bb[L], hin, agg, nullptr, nullptr);
        }
    }

    // ---- fc: bn(relu(x4 @ fcW + fcB)) ----
    fill(ssum, 256, 0.f);
    sage_gemm<128, 64, false, true, true><<<GB, 128, 0, stream>>>(
        agg, nullptr, fcW, nullptr, nullptr, fcB, nullptr, hB, ssum, ssq);
    bn_finalize_kernel<<<1, 64, 0, stream>>>(ssum, ssq, g[3], be[3], sc, sh, 64);
    bn_apply_kernel<<<(int)((NC2 + 255) / 256), 256, 0, stream>>>(hB, sc, sh, hA, NC2, 63);

    // ---- fc1: bn(relu(h @ fc1W + fc1B)) ----
    fill(ssum, 256, 0.f);
    sage_gemm<64, 64, false, true, true><<<GB, 128, 0, stream>>>(
        hA, nullptr, fc1W, nullptr, nullptr, fc1B, nullptr, hB, ssum, ssq);
    bn_finalize_kernel<<<1, 64, 0, stream>>>(ssum, ssq, g[4], be[4], sc, sh, 64);
    bn_apply_kernel<<<(int)((NC2 + 255) / 256), 256, 0, stream>>>(hB, sc, sh, hA, NC2, 63);

    // ---- fc2: h @ fc2W + fc2B -> d_out ----
    sage_gemm<64, 64, false, false, false><<<GB, 128, 0, stream>>>(
        hA, nullptr, fc2W, nullptr, nullptr, fc2B, nullptr, (float*)d_out, nullptr, nullptr);
}